// GCN_6193342841082
// MI455X (gfx1250) — compile-verified
//
#include <hip/hip_runtime.h>

typedef __attribute__((ext_vector_type(2))) float v2f;
typedef __attribute__((ext_vector_type(8))) float v8f;

#define DIM 128

__device__ __forceinline__ v8f wmma_f32_16x16x4(v2f a, v2f b, v8f c) {
  return __builtin_amdgcn_wmma_f32_16x16x4_f32(false, a, false, b, (short)0, c,
                                               false, false);
}

// ---------------------------------------------------------------------------
// Edge tile body: one wave, 16 edges. Split into src-half / dst-half so every
// global access is base + immediate offset (no per-iteration address VALU),
// and templated on FULL so the common case has unpredicated NT stores.
// ---------------------------------------------------------------------------
template <bool FULL>
__device__ __forceinline__ void edge_tile_body(const float* __restrict__ srow,
                                               const float* __restrict__ drow,
                                               float* __restrict__ orow,
                                               const v2f* __restrict__ BtabLane,
                                               v8f& acc, bool wr, int coff) {
#pragma unroll 8
  for (int kk = 0; kk < 32; ++kk) {
    const int c = 4 * kk + coff;
    const v2f a = *(const v2f*)(srow + c);
    const v2f b = BtabLane[kk * 32];
    acc = wmma_f32_16x16x4(a, b, acc);
    if (FULL || wr) __builtin_nontemporal_store(a, (v2f*)(orow + c));
  }
#pragma unroll 8
  for (int kk = 32; kk < 64; ++kk) {
    const int c = 4 * kk + coff;
    const v2f a = *(const v2f*)(drow + (c - DIM));
    const v2f b = BtabLane[kk * 32];
    acc = wmma_f32_16x16x4(a, b, acc);
    if (FULL || wr) __builtin_nontemporal_store(a, (v2f*)(orow + c));
  }
}

// edge_emb[e][0:256] = concat(x[src[e]], x[dst[e]])  (640 MB NT stream-out)
// edge_x[e][0:2]     = edge_emb[e] @ We + be         (fused WMMA f32 16x16x4)
// A-matrix f32 16x4 layout: lane l<16 -> row l, K={0,1}; lane l>=16 -> row
// l-16, K={2,3}; each lane's float2 gather chunk IS its A fragment.
__global__ __launch_bounds__(256) void gcn_edge_kernel(
    const float* __restrict__ x, const int* __restrict__ eidx,
    const float* __restrict__ We, const float* __restrict__ be,
    float* __restrict__ edge_emb, float* __restrict__ edge_x, int E) {
  // Pre-swizzled B fragments: Btab[kk][lane] = {B_vgpr0, B_vgpr1}.
  // B is 4x16 (K x N): cols 0..1 = We, cols 2..15 = 0.
  // vgpr0 holds K = 2*(lane>>4), vgpr1 holds K+1, N = lane&15.
  __shared__ v2f Btab[64 * 32];  // 16 KB
  const int tid = threadIdx.x;
  for (int i = tid; i < 64 * 32; i += blockDim.x) {
    const int kk = i >> 5, l = i & 31;
    const int n = l & 15;
    const int k0 = 4 * kk + 2 * (l >> 4);
    v2f b;
    b.x = (n < 2) ? We[k0 * 2 + n] : 0.0f;
    b.y = (n < 2) ? We[(k0 + 1) * 2 + n] : 0.0f;
    Btab[i] = b;
  }
  __syncthreads();

  const int lane = tid & 31;
  const int wave = tid >> 5;
  const long tile = (long)blockIdx.x * (blockDim.x >> 5) + wave;  // 16 edges
  const long e0 = tile * 16;
  if (e0 >= E) return;  // uniform per wave

  const int m = lane & 15;
  const long e = e0 + m;
  const long ec = (e < E) ? e : (long)(E - 1);  // clamp: EXEC all-1 for WMMA
  const int src = eidx[ec];
  const int dst = eidx[(long)E + ec];
  const float* __restrict__ srow = x + (long)src * DIM;
  const float* __restrict__ drow = x + (long)dst * DIM;
  const int coff = 2 * (lane >> 4);  // 0 or 2
  float* __restrict__ orow = edge_emb + e * 256;
  const v2f* __restrict__ BtabLane = Btab + lane;

  v8f acc = {};
  if (e0 + 16 <= E) {  // common case: whole tile in range, no predication
    edge_tile_body<true>(srow, drow, orow, BtabLane, acc, true, coff);
  } else {  // single trailing tile
    edge_tile_body<false>(srow, drow, orow, BtabLane, acc, e < E, coff);
  }

  // D layout: VGPR v, lanes 0-15 -> D[v][lane]; lanes 16-31 -> D[v+8][lane-16]
  // Only N=0,1 are real outputs -> lanes {0,1,16,17} hold edge_x.
  if ((lane & 15) < 2) {
    const int col = lane & 1;
    const float bias = be[col];
    const int mbase = (lane >> 4) * 8;
#pragma unroll
    for (int v = 0; v < 8; ++v) {
      const long ee = e0 + mbase + v;
      if (ee < E) edge_x[ee * 2 + col] = acc[v] + bias;
    }
  }
}

// ---------------------------------------------------------------------------
// Node tile body + kernel: node_emb = x (NT copy); node_x = x @ Wn + bn.
// ---------------------------------------------------------------------------
template <bool FULL>
__device__ __forceinline__ void node_tile_body(const float* __restrict__ row,
                                               float* __restrict__ orow,
                                               const v2f* __restrict__ BtabLane,
                                               v8f& acc, bool wr, int coff) {
#pragma unroll 8
  for (int kk = 0; kk < 32; ++kk) {
    const int c = 4 * kk + coff;
    const v2f a = *(const v2f*)(row + c);
    const v2f b = BtabLane[kk * 32];
    acc = wmma_f32_16x16x4(a, b, acc);
    if (FULL || wr) __builtin_nontemporal_store(a, (v2f*)(orow + c));
  }
}

__global__ __launch_bounds__(256) void gcn_node_kernel(
    const float* __restrict__ x, const float* __restrict__ Wn,
    const float* __restrict__ bn, float* __restrict__ node_emb,
    float* __restrict__ node_x, int N) {
  __shared__ v2f Btab[32 * 32];  // 8 KB
  const int tid = threadIdx.x;
  for (int i = tid; i < 32 * 32; i += blockDim.x) {
    const int kk = i >> 5, l = i & 31;
    const int n = l & 15;
    const int k0 = 4 * kk + 2 * (l >> 4);
    v2f b;
    b.x = (n < 2) ? Wn[k0 * 2 + n] : 0.0f;
    b.y = (n < 2) ? Wn[(k0 + 1) * 2 + n] : 0.0f;
    Btab[i] = b;
  }
  __syncthreads();

  const int lane = tid & 31;
  const int wave = tid >> 5;
  const long tile = (long)blockIdx.x * (blockDim.x >> 5) + wave;
  const long r0 = tile * 16;
  if (r0 >= N) return;

  const int m = lane & 15;
  const long r = r0 + m;
  const long rc = (r < N) ? r : (long)(N - 1);
  const float* __restrict__ row = x + rc * DIM;
  const int coff = 2 * (lane >> 4);
  float* __restrict__ orow = node_emb + r * DIM;
  const v2f* __restrict__ BtabLane = Btab + lane;

  v8f acc = {};
  if (r0 + 16 <= N) {
    node_tile_body<true>(row, orow, BtabLane, acc, true, coff);
  } else {
    node_tile_body<false>(row, orow, BtabLane, acc, r < N, coff);
  }

  if ((lane & 15) < 2) {
    const int col = lane & 1;
    const float bias = bn[col];
    const int mbase = (lane >> 4) * 8;
#pragma unroll
    for (int v = 0; v < 8; ++v) {
      const long rr = r0 + mbase + v;
      if (rr < N) node_x[rr * 2 + col] = acc[v] + bias;
    }
  }
}

extern "C" void kernel_launch(void* const* d_in, const int* in_sizes, int n_in,
                              void* d_out, int out_size, void* d_ws,
                              size_t ws_size, hipStream_t stream) {
  const float* x = (const float*)d_in[0];   // [N,128] f32
  const int* eidx = (const int*)d_in[1];    // [2,E] int32
  const float* Wn = (const float*)d_in[2];  // [128,2]
  const float* bn = (const float*)d_in[3];  // [2]
  const float* We = (const float*)d_in[4];  // [256,2]
  const float* be = (const float*)d_in[5];  // [2]

  const int N = in_sizes[0] / DIM;  // 50000
  const int E = in_sizes[1] / 2;    // 625000

  // Output tuple, concatenated flat in return order.
  float* out = (float*)d_out;
  float* node_emb = out;                         // N*128
  float* edge_emb = node_emb + (long)N * DIM;    // E*256
  float* node_x = edge_emb + (long)E * 2 * DIM;  // N*2
  float* edge_x = node_x + (long)N * 2;          // E*2

  const int threads = 256;
  const int tiles_per_block = threads / 32;  // 8 wave-tiles (16 rows each)

  const int node_tiles = (N + 15) / 16;
  const int node_blocks = (node_tiles + tiles_per_block - 1) / tiles_per_block;
  gcn_node_kernel<<<node_blocks, threads, 0, stream>>>(x, Wn, bn, node_emb,
                                                       node_x, N);

  const int edge_tiles = (E + 15) / 16;
  const int edge_blocks = (edge_tiles + tiles_per_block - 1) / tiles_per_block;
  gcn_edge_kernel<<<edge_blocks, threads, 0, stream>>>(x, eidx, We, be,
                                                       edge_emb, edge_x, E);
}